// DecodePredictions_1846835937424
// MI455X (gfx1250) — compile-verified
//
#include <hip/hip_runtime.h>
#include <math.h>

#define N_ANCHORS   200000
#define NUM_CLASSES 91
#define ROW         95      // 4 loc + 91 cls
#define IMG_LIM     512.0f
#define SCORE_THR   0.02f
#define IOU_THR     0.5f
#define MAX_DET     100

typedef float v2f __attribute__((ext_vector_type(2)));
typedef float v8f __attribute__((ext_vector_type(8)));

#define DEC_TILE    128     // anchors per workgroup (8 waves * 16)
#define DEC_THREADS 256

#define NMS_BLOCKS  64
#define NMS_THREADS 1024
#define CHUNK       3125    // 64 * 3125 == 200000 exactly

__device__ __forceinline__ void argbetter(float& bv, int& bi, float v, int i) {
    if (v > bv || (v == bv && i < bi)) { bv = v; bi = i; }
}

__global__ void init_kernel(int* counter) {
    if (threadIdx.x == 0) *counter = 0;
}

// ---------------------------------------------------------------------------
// Kernel 1: per-anchor softmax argmax + score + box decode.
// One streaming pass over predictions (76 MB), staged into LDS with the CDNA5
// async DMA path (GLOBAL_LOAD_ASYNC_TO_LDS_B128, ASYNCcnt). Sum-of-exp is a
// 16x91 x ones GEMV chunked through V_WMMA_F32_16X16X4_F32.
// ---------------------------------------------------------------------------
__global__ __launch_bounds__(DEC_THREADS)
void decode_kernel(const float* __restrict__ preds,
                   const float* __restrict__ dflt,
                   float4* __restrict__ oBox,
                   float*  __restrict__ oScore,
                   int*    __restrict__ oId)
{
    __shared__ float tile[DEC_TILE * ROW];          // 48,640 B (16B aligned size)
    const int tileBase = blockIdx.x * DEC_TILE;
    const long long gBase = (long long)tileBase * ROW;   // multiple of 4 dwords
    const long long total = (long long)N_ANCHORS * ROW;  // 19,000,000 (mult of 4)

    // ---- stage tile via async global->LDS DMA (no VGPR round-trip) --------
    {
        const uint32_t ldsBase = (uint32_t)(size_t)(void*)tile;
        const int nVec = (DEC_TILE * ROW) / 4;      // 3040 b128 transfers
        const long long totVec = total / 4;
        for (int i = threadIdx.x; i < nVec; i += DEC_THREADS) {
            const long long gv = (gBase >> 2) + i;
            if (gv < totVec) {                      // per-lane EXEC predication
                const float*  gp = preds + (size_t)gv * 4;
                const uint32_t la = ldsBase + (uint32_t)i * 16u;
                asm volatile("global_load_async_to_lds_b128 %0, %1, off"
                             :: "v"(la), "v"(gp) : "memory");
            }
        }
        asm volatile("s_wait_asynccnt 0x0" ::: "memory");
    }
    __syncthreads();

    const int lane = threadIdx.x & 31;
    const int wave = threadIdx.x >> 5;
    const int m    = lane & 15;                      // anchor within wave's 16
    const int half = lane >> 4;                      // lane-pair half
    const int aLoc = wave * 16 + m;
    const int aGlb = tileBase + aLoc;
    const float* row = &tile[aLoc * ROW];            // stride 95 -> conflict-free

    // ---- max / argmax over 91 logits: lane pair splits the row 46/45 ----
    float mx = -INFINITY; int amax = 0;
    const int c0 = half ? 46 : 0;
    const int c1 = half ? NUM_CLASSES : 46;
    for (int c = c0; c < c1; ++c) {
        float v = row[4 + c];
        if (v > mx) { mx = v; amax = c; }
    }
    {
        float mx2 = __shfl_xor(mx, 16);
        int   am2 = __shfl_xor(amax, 16);
        if (mx2 > mx || (mx2 == mx && am2 < amax)) { mx = mx2; amax = am2; }
    }

    // ---- sum(exp(logit-max)) via WMMA f32 16x16x4 with a ones B-matrix ----
    // A layout (ISA 7.12.2, 32-bit A 16x4): v0 = K=0 (lanes 0-15) / K=2
    // (lanes 16-31); v1 = K=1 / K=3.  24 chunks cover K=0..95 (pad >= 91).
    v8f acc = {};
    v2f ones; ones[0] = 1.0f; ones[1] = 1.0f;
    #pragma unroll
    for (int kb = 0; kb < 96; kb += 4) {
        const int k0 = kb + half * 2;
        const int i0 = (k0     < NUM_CLASSES) ? k0     : 0;   // clamp LDS index
        const int i1 = (k0 + 1 < NUM_CLASSES) ? k0 + 1 : 0;
        v2f a;
        a[0] = (k0     < NUM_CLASSES) ? __expf(row[4 + i0] - mx) : 0.0f;
        a[1] = (k0 + 1 < NUM_CLASSES) ? __expf(row[4 + i1] - mx) : 0.0f;
        acc = __builtin_amdgcn_wmma_f32_16x16x4_f32(false, a, false, ones,
                                                    (short)0, acc, false, false);
    }

    // ---- extract row-sum for this lane's anchor from D (M=r+8*hi, N=l%16) --
    float sum = 0.0f;
    const int srcLane = (lane < 8) ? lane : (lane + 16);
    #pragma unroll
    for (int r = 0; r < 8; ++r) {
        float v = __shfl(acc[r], srcLane);
        if ((lane & 7) == r) sum = v;
    }

    if (lane < 16 && aGlb < N_ANCHORS) {
        const float score = 1.0f / sum;              // exp(mx-mx)/sumexp
        const bool  keep  = (amax != 0) && (score >= SCORE_THR);
        const float4 db = ((const float4*)dflt)[aGlb];   // cx, cy, w, h
        const float l0 = row[0] * 0.1f, l1 = row[1] * 0.1f;
        const float l2 = row[2] * 0.2f, l3 = row[3] * 0.2f;
        const float cx = l0 * db.z + db.x;
        const float cy = l1 * db.w + db.y;
        const float w  = __expf(l2) * db.z;
        const float h  = __expf(l3) * db.w;
        float4 bx;
        bx.x = fminf(fmaxf(cx - 0.5f * w, 0.0f), IMG_LIM);
        bx.y = fminf(fmaxf(cy - 0.5f * h, 0.0f), IMG_LIM);
        bx.z = fminf(fmaxf(cx + 0.5f * w, 0.0f), IMG_LIM);
        bx.w = fminf(fmaxf(cy + 0.5f * h, 0.0f), IMG_LIM);
        oBox[aGlb]   = bx;
        oScore[aGlb] = keep ? score : -INFINITY;
        oId[aGlb]    = amax;
    }
}

// ---------------------------------------------------------------------------
// Kernel 2: persistent greedy NMS. 64 blocks pin their candidate chunk in LDS
// (62.5 KB of the 320 KB/WGP); per round: fused suppress+rescan from LDS,
// agent-scope atomic grid barriers, block 0 selects winner & writes outputs.
// ---------------------------------------------------------------------------
__global__ __launch_bounds__(NMS_THREADS)
void nms_kernel(const float4* __restrict__ boxes,
                const float*  __restrict__ scores,
                const int*    __restrict__ ids,
                float2* partials,      // [64] (bestScore, idx-bits)
                float*  selRec,        // [8]  box4, valid-bits, idx-bits
                int*    counter,
                float*  __restrict__ out)
{
    __shared__ float  sSc[CHUNK];                    // 12,500 B
    __shared__ float4 sBx[CHUNK];                    // 50,000 B
    __shared__ float  rv[NMS_THREADS / 32];
    __shared__ int    ri[NMS_THREADS / 32];

    const int tid   = threadIdx.x;
    const int lane  = tid & 31;
    const int wid   = tid >> 5;
    const int blk   = blockIdx.x;
    const int start = blk * CHUNK;

    for (int i = tid; i < CHUNK; i += NMS_THREADS) {
        const int g = start + i;
        if (g < N_ANCHORS) { sSc[i] = scores[g]; sBx[i] = boxes[g]; }
        else               { sSc[i] = -INFINITY; sBx[i] = make_float4(0,0,0,0); }
    }
    __syncthreads();

    int phase = 0;  // only tid 0's copy matters

    auto gridBar = [&]() {
        __syncthreads();
        if (tid == 0) {
            __hip_atomic_fetch_add(counter, 1, __ATOMIC_RELEASE,
                                   __HIP_MEMORY_SCOPE_AGENT);
            ++phase;
            while (__hip_atomic_load(counter, __ATOMIC_ACQUIRE,
                                     __HIP_MEMORY_SCOPE_AGENT) < phase * NMS_BLOCKS)
                __builtin_amdgcn_s_sleep(1);
        }
        __syncthreads();
    };

    auto blockReduce = [&](float bv, int bi) {
        #pragma unroll
        for (int off = 16; off; off >>= 1) {
            float v2 = __shfl_xor(bv, off);
            int   i2 = __shfl_xor(bi, off);
            argbetter(bv, bi, v2, i2);
        }
        if (lane == 0) { rv[wid] = bv; ri[wid] = bi; }
        __syncthreads();
        if (wid == 0) {
            bv = rv[lane]; bi = ri[lane];
            #pragma unroll
            for (int off = 16; off; off >>= 1) {
                float v2 = __shfl_xor(bv, off);
                int   i2 = __shfl_xor(bi, off);
                argbetter(bv, bi, v2, i2);
            }
            if (lane == 0) partials[blk] = make_float2(bv, __int_as_float(bi));
        }
        __syncthreads();
    };

    // initial per-block argmax
    {
        float bv = -INFINITY; int bi = 0x7fffffff;
        for (int i = tid; i < CHUNK; i += NMS_THREADS)
            argbetter(bv, bi, sSc[i], start + i);
        blockReduce(bv, bi);
    }

    for (int it = 0; it < MAX_DET; ++it) {
        gridBar();                                   // partials visible

        if (blk == 0 && wid == 0) {                  // global select + emit
            float v1 = partials[lane].x;
            int   i1 = __float_as_int(partials[lane].y);
            float v2 = partials[lane + 32].x;
            int   i2 = __float_as_int(partials[lane + 32].y);
            argbetter(v1, i1, v2, i2);
            #pragma unroll
            for (int off = 16; off; off >>= 1) {
                float vv = __shfl_xor(v1, off);
                int   ii = __shfl_xor(i1, off);
                argbetter(v1, i1, vv, ii);
            }
            if (lane == 0) {
                const bool ok = v1 > -INFINITY;
                const float4 wb = ok ? boxes[i1] : make_float4(0,0,0,0);
                const int    wi = ok ? ids[i1]   : 0;
                out[it * 4 + 0] = wb.x; out[it * 4 + 1] = wb.y;
                out[it * 4 + 2] = wb.z; out[it * 4 + 3] = wb.w;
                out[4 * MAX_DET + it] = ok ? (float)(wi - 1) : -1.0f;
                out[5 * MAX_DET + it] = ok ? v1 : 0.0f;
                selRec[0] = wb.x; selRec[1] = wb.y;
                selRec[2] = wb.z; selRec[3] = wb.w;
                selRec[4] = __int_as_float(ok ? 1 : 0);
                selRec[5] = __int_as_float(i1);
            }
        }

        gridBar();                                   // selection visible

        const float4 sb  = *(const float4*)selRec;   // broadcast via L2
        const int sval   = __float_as_int(selRec[4]);
        const int sidx   = __float_as_int(selRec[5]);
        const float sArea = (sb.z - sb.x) * (sb.w - sb.y);

        float bv = -INFINITY; int bi = 0x7fffffff;
        for (int i = tid; i < CHUNK; i += NMS_THREADS) {
            const int g = start + i;
            float v = sSc[i];
            if (sval && v > -INFINITY) {
                const float4 bb = sBx[i];
                const float ix1 = fmaxf(sb.x, bb.x), iy1 = fmaxf(sb.y, bb.y);
                const float ix2 = fminf(sb.z, bb.z), iy2 = fminf(sb.w, bb.w);
                const float iw = fmaxf(ix2 - ix1, 0.0f);
                const float ih = fmaxf(iy2 - iy1, 0.0f);
                const float inter = iw * ih;
                const float uni = sArea + (bb.z - bb.x) * (bb.w - bb.y) - inter + 1e-9f;
                if (inter / uni > IOU_THR || g == sidx) { v = -INFINITY; sSc[i] = v; }
            }
            argbetter(bv, bi, v, g);
        }
        blockReduce(bv, bi);                         // partials for round it+1
    }
}

// ---------------------------------------------------------------------------
extern "C" void kernel_launch(void* const* d_in, const int* in_sizes, int n_in,
                              void* d_out, int out_size, void* d_ws, size_t ws_size,
                              hipStream_t stream)
{
    const float* preds = (const float*)d_in[0];   // [1, 200000, 95] f32
    const float* dflt  = (const float*)d_in[1];   // [200000, 4] f32
    float* out = (float*)d_out;                   // 400 boxes + 100 ids + 100 scores

    char* ws = (char*)d_ws;
    float4* wBox   = (float4*)(ws);                                  // 3.2 MB
    float*  wScore = (float*)(ws + (size_t)16 * N_ANCHORS);          // 0.8 MB
    int*    wId    = (int*)  (ws + (size_t)20 * N_ANCHORS);          // 0.8 MB
    float2* wPart  = (float2*)(ws + (size_t)24 * N_ANCHORS);         // 512 B
    float*  wSel   = (float*)(ws + (size_t)24 * N_ANCHORS + 1024);   // 32 B
    int*    wCnt   = (int*)  (ws + (size_t)24 * N_ANCHORS + 1152);   // 4 B

    init_kernel<<<1, 32, 0, stream>>>(wCnt);
    decode_kernel<<<(N_ANCHORS + DEC_TILE - 1) / DEC_TILE, DEC_THREADS, 0, stream>>>(
        preds, dflt, wBox, wScore, wId);
    nms_kernel<<<NMS_BLOCKS, NMS_THREADS, 0, stream>>>(
        wBox, wScore, wId, wPart, wSel, wCnt, out);
}